// GradientAwareAreaUtil_5531917877519
// MI455X (gfx1250) — compile-verified
//
#include <hip/hip_runtime.h>

#define IMG_H 256
#define IMG_W 256
#define ROWS_PER_CHUNK 16
#define NCHUNK (IMG_H / ROWS_PER_CHUNK)          // 16
#define TPB 256
#define CHUNK_FLOATS (ROWS_PER_CHUNK * IMG_W)    // 4096 floats = 16 KB
#define SWEEPS (CHUNK_FLOATS / (TPB * 4))        // 4 x b128 per lane per chunk
#define PENALTY 0.1f

__global__ __launch_bounds__(TPB) void tv_area_kernel(const float* __restrict__ Y,
                                                      float* __restrict__ out)
{
    __shared__ __align__(16) float buf[2][ROWS_PER_CHUNK][IMG_W];
    __shared__ __align__(16) float lastrow[IMG_W];
    __shared__ float red[TPB / 32];

    const int b = blockIdx.x;
    const int t = threadIdx.x;
    const float* __restrict__ img = Y + (size_t)b * (IMG_H * IMG_W);

    // 32-bit LDS byte offsets (flat addr low 32 bits == LDS offset per ISA 10.2)
    const unsigned lds_buf0 = (unsigned)(size_t)&buf[0][0][0];
    const unsigned lds_buf1 = (unsigned)(size_t)&buf[1][0][0];

    // Issue one chunk: 4 async b128 copies per lane, whole block covers 16 KB.
    auto issue_chunk = [&](int k, unsigned lds_base) {
        unsigned goff = (unsigned)(k * CHUNK_FLOATS * 4) + (unsigned)t * 16u;
        unsigned loff = lds_base + (unsigned)t * 16u;
#pragma unroll
        for (int i = 0; i < SWEEPS; ++i) {
            asm volatile("global_load_async_to_lds_b128 %0, %1, %2"
                         :: "v"(loff + (unsigned)i * (TPB * 16u)),
                            "v"(goff + (unsigned)i * (TPB * 16u)),
                            "s"(img)
                         : "memory");
        }
    };

    issue_chunk(0, lds_buf0);

    float area = 0.0f;
    float grad = 0.0f;

    for (int k = 0; k < NCHUNK; ++k) {
        float (* __restrict__ bufc)[IMG_W] = buf[k & 1];

        if (k + 1 < NCHUNK) {
            // prefetch next chunk into the other buffer, then wait for current
            issue_chunk(k + 1, (k & 1) ? lds_buf0 : lds_buf1);
            asm volatile("s_wait_asynccnt 4" ::: "memory");
        } else {
            asm volatile("s_wait_asynccnt 0" ::: "memory");
        }
        __syncthreads();   // chunk k visible to all waves

        // 1024 float4 positions per chunk, 4 per thread
#pragma unroll
        for (int i = 0; i < 4; ++i) {
            const int j = t + TPB * i;
            const int g = j >> 6;        // row within chunk
            const int q = j & 63;        // float4 column
            const float4 v = ((const float4*)&bufc[g][0])[q];

            area += (v.x + v.y) + (v.z + v.w);

            float hd = fabsf(v.x - v.y) + fabsf(v.y - v.z) + fabsf(v.z - v.w);
            if (q < 63) hd += fabsf(v.w - bufc[g][4 * q + 4]);

            float vd = 0.0f;
            if (g > 0) {
                const float4 u = ((const float4*)&bufc[g - 1][0])[q];
                vd = fabsf(v.x - u.x) + fabsf(v.y - u.y) +
                     fabsf(v.z - u.z) + fabsf(v.w - u.w);
            } else if (k > 0) {
                const float4 u = ((const float4*)&lastrow[0])[q];
                vd = fabsf(v.x - u.x) + fabsf(v.y - u.y) +
                     fabsf(v.z - u.z) + fabsf(v.w - u.w);
            }
            grad += hd + vd;
        }
        __syncthreads();   // everyone done reading lastrow + bufc

        // persist this chunk's last row for the next chunk's top-row vdiff
        if (t < 64) {
            ((float4*)&lastrow[0])[t] = ((const float4*)&bufc[ROWS_PER_CHUNK - 1][0])[t];
        }
        __syncthreads();   // lastrow written before next iteration issues/reads
    }

    // out = area - 0.1 * grad; linear, so reduce a single value
    float val = area - PENALTY * grad;

    // wave32 butterfly reduce
#pragma unroll
    for (int o = 16; o > 0; o >>= 1) val += __shfl_xor(val, o, 32);

    if ((t & 31) == 0) red[t >> 5] = val;
    __syncthreads();
    if (t == 0) {
        float s = 0.0f;
#pragma unroll
        for (int w = 0; w < TPB / 32; ++w) s += red[w];
        out[b] = s;
    }
}

extern "C" void kernel_launch(void* const* d_in, const int* in_sizes, int n_in,
                              void* d_out, int out_size, void* d_ws, size_t ws_size,
                              hipStream_t stream)
{
    (void)n_in; (void)d_ws; (void)ws_size; (void)out_size;
    const float* Y = (const float*)d_in[0];
    float* out = (float*)d_out;
    const int batch = in_sizes[0] / (IMG_H * IMG_W);   // 512
    tv_area_kernel<<<batch, TPB, 0, stream>>>(Y, out);
}